// GraphSAGE_82875688943940
// MI455X (gfx1250) — compile-verified
//
#include <hip/hip_runtime.h>

// ---------------- problem constants ----------------
#define NS 100000
#define NC 5000
#define NK 20000
#define E_ENR 500000
#define E_ENG 500000
#define E_PRQ 50000
#define E_COV 100000
#define E_CPR 100000
#define CDIM 128

typedef __attribute__((ext_vector_type(2))) float v2f;
typedef __attribute__((ext_vector_type(8))) float v8f;

// ---------------- small helpers ----------------
__device__ __forceinline__ float leaky02(float a) { return a > 0.f ? a : 0.2f * a; }

// order-preserving float -> uint encoding for atomicMax-based segment max
__device__ __forceinline__ unsigned fenc(float f) {
  unsigned u = __float_as_uint(f);
  return (u & 0x80000000u) ? ~u : (u | 0x80000000u);
}
__device__ __forceinline__ float fdec(unsigned u) {
  unsigned b = (u & 0x80000000u) ? (u ^ 0x80000000u) : ~u;
  return __uint_as_float(b);
}
// enc==0 (init) decodes to -NaN -> non-finite -> treated as empty segment (m = 0),
// matching the reference's isfinite() replacement.

// ---------------- small-parameter scratch layout (floats, within sp[]) ----------------
#define SP_WSAS_ENR 0     // 5
#define SP_WSAS_ENG 8     // 5
#define SP_WDAD_RENR 16   // 5
#define SP_WDAD_RENG 24   // 5
#define SP_WSAS_RENR 32   // 64
#define SP_WDAD_ENR 96    // 64
#define SP_WSAS_RENG 160  // 32
#define SP_WDAD_ENG 192   // 32
#define SP_CE 224         // 4: [enr, eng, renr, reng]
#define SP_BIAS_COURSE 256
#define SP_BIAS_CONCEPT 384
#define SP_BIAS_SOUT 512
#define SP_SBIAS 640
#define SP_TOTAL 1024

struct PrepArgs {
  const float *Ws_enr, *as_enr, *Ws_eng, *as_eng;
  const float *Wd_renr, *ad_renr, *Wd_reng, *ad_reng;
  const float *Ws_renr, *as_renr, *Wd_enr, *ad_enr;
  const float *Ws_reng, *as_reng, *Wd_eng, *ad_eng;
  const float *We_enr, *ae_enr, *We_eng, *ae_eng;
  const float *We_renr, *ae_renr, *We_reng, *ae_reng;
  const float *b_enr, *b_eng, *b_renr, *b_reng;
  const float *bl_prq, *bl_cov, *bl_cpr, *bl2_renr, *bl2_reng, *proj_b;
};

// One block, 256 threads: all tiny derived vectors + folded biases.
__global__ void prep_vectors_kernel(PrepArgs p, float* __restrict__ sp) {
  int t = threadIdx.x;
  const float* row = nullptr;
  const float* vec = nullptr;
  if (t < 5)                 { row = p.Ws_enr  + t * 128;         vec = p.as_enr; }
  else if (t >= 8  && t < 13){ row = p.Ws_eng  + (t - 8) * 128;   vec = p.as_eng; }
  else if (t >= 16 && t < 21){ row = p.Wd_renr + (t - 16) * 128;  vec = p.ad_renr; }
  else if (t >= 24 && t < 29){ row = p.Wd_reng + (t - 24) * 128;  vec = p.ad_reng; }
  else if (t >= 32 && t < 96){ row = p.Ws_renr + (t - 32) * 128;  vec = p.as_renr; }
  else if (t >= 96 && t < 160){row = p.Wd_enr  + (t - 96) * 128;  vec = p.ad_enr; }
  else if (t >= 160 && t < 192){row = p.Ws_reng + (t - 160) * 128; vec = p.as_reng; }
  else if (t >= 192 && t < 224){row = p.Wd_eng  + (t - 192) * 128; vec = p.ad_eng; }
  else if (t == 224) { row = p.We_enr;  vec = p.ae_enr; }
  else if (t == 225) { row = p.We_eng;  vec = p.ae_eng; }
  else if (t == 226) { row = p.We_renr; vec = p.ae_renr; }
  else if (t == 227) { row = p.We_reng; vec = p.ae_reng; }
  if (row) {
    float acc = 0.f;
    for (int i = 0; i < 128; ++i) acc = fmaf(row[i], vec[i], acc);
    sp[t] = acc;  // job index == sp offset by construction
  }
  if (t < 128) {
    sp[SP_BIAS_COURSE + t]  = p.b_enr[t] + p.bl_prq[t];
    sp[SP_BIAS_CONCEPT + t] = p.b_eng[t] + p.bl_cov[t] + p.bl_cpr[t];
    sp[SP_BIAS_SOUT + t]    = p.bl2_renr[t] + p.bl2_reng[t];
    sp[SP_SBIAS + t]        = 0.3f * p.proj_b[t] + p.b_renr[t] + p.b_reng[t];
  }
}

__global__ void add_arrays_kernel(const float* __restrict__ a, const float* __restrict__ b,
                                  float* __restrict__ o, int n) {
  int i = blockIdx.x * blockDim.x + threadIdx.x;
  if (i < n) o[i] = a[i] + b[i];
}

// per-node attention scalar tables: o1 = x . v1, o2 = x . v2 (row-wise)
__global__ void node_scores_kernel(const float* __restrict__ x, int F,
                                   const float* __restrict__ v1, const float* __restrict__ v2,
                                   float* __restrict__ o1, float* __restrict__ o2, int N) {
  int n = blockIdx.x * blockDim.x + threadIdx.x;
  if (n >= N) return;
  const float* xp = x + (size_t)n * F;
  float a = 0.f, b = 0.f;
  for (int i = 0; i < F; ++i) { float xv = xp[i]; a = fmaf(xv, v1[i], a); b = fmaf(xv, v2[i], b); }
  o1[n] = a; o2[n] = b;
}

// s_acc init: 0.3*(x_student @ proj_W) + folded bias (0.3*proj_b + b_gat_renr + b_gat_reng)
__global__ void s_skip_kernel(const float* __restrict__ xs, const float* __restrict__ projW,
                              const float* __restrict__ sbias, float* __restrict__ sacc, int N) {
  int idx = blockIdx.x * blockDim.x + threadIdx.x;
  if (idx >= N * CDIM) return;
  int n = idx >> 7, c = idx & 127;
  const float* xp = xs + (size_t)n * 5;
  float v = xp[0] * projW[c] + xp[1] * projW[128 + c] + xp[2] * projW[256 + c] +
            xp[3] * projW[384 + c] + xp[4] * projW[512 + c];
  sacc[idx] = 0.3f * v + sbias[c];
}

// ---------------- GAT edge kernels ----------------
__global__ void gat_edge_max_kernel(const int* __restrict__ src, const int* __restrict__ dst,
                                    const float* __restrict__ ew, const float* __restrict__ cep,
                                    const float* __restrict__ ssc, const float* __restrict__ dsc,
                                    unsigned* __restrict__ menc, int E) {
  int e = blockIdx.x * blockDim.x + threadIdx.x;
  if (e >= E) return;
  int s = src[e], d = dst[e];
  float a = leaky02(ssc[s] + dsc[d] + ew[e] * cep[0]);
  atomicMax(&menc[d], fenc(a));
}

__global__ void gat_edge_den_kernel(const int* __restrict__ src, const int* __restrict__ dst,
                                    const float* __restrict__ ew, const float* __restrict__ cep,
                                    const float* __restrict__ ssc, const float* __restrict__ dsc,
                                    const unsigned* __restrict__ menc, float* __restrict__ den, int E) {
  int e = blockIdx.x * blockDim.x + threadIdx.x;
  if (e >= E) return;
  int s = src[e], d = dst[e];
  float a = leaky02(ssc[s] + dsc[d] + ew[e] * cep[0]);
  float m = fdec(menc[d]);
  if (!(m > -1e38f && m < 1e38f)) m = 0.f;
  atomicAdd(&den[d], expf(a - m));
}

__device__ __forceinline__ float edge_alpha(int s, int d, float ewv, float ce,
                                            const float* ssc, const float* dsc,
                                            const unsigned* menc, const float* den) {
  float a = leaky02(ssc[s] + dsc[d] + ewv * ce);
  float m = fdec(menc[d]);
  if (!(m > -1e38f && m < 1e38f)) m = 0.f;
  return expf(a - m) / (den[d] + 1e-16f);
}

// wave32 per edge: gather 128-wide hs[src] (float4), scatter alpha*hs into out[dst]
__global__ void gat_edge_agg_mat_kernel(const int* __restrict__ src, const int* __restrict__ dst,
                                        const float* __restrict__ ew, const float* __restrict__ cep,
                                        const float* __restrict__ ssc, const float* __restrict__ dsc,
                                        const unsigned* __restrict__ menc, const float* __restrict__ den,
                                        const float* __restrict__ hs, float* __restrict__ out, int E) {
  int gid = blockIdx.x * blockDim.x + threadIdx.x;
  int e = gid >> 5;
  if (e >= E) return;
  int lane = gid & 31;
  int s = src[e], d = dst[e];
  float alpha = edge_alpha(s, d, ew[e], cep[0], ssc, dsc, menc, den);
  float4 h = *(const float4*)(hs + (size_t)s * CDIM + lane * 4);
  float* op = out + (size_t)d * CDIM + lane * 4;
  atomicAdd(op + 0, alpha * h.x);
  atomicAdd(op + 1, alpha * h.y);
  atomicAdd(op + 2, alpha * h.z);
  atomicAdd(op + 3, alpha * h.w);
}

// wave32 per edge: hs row computed on the fly from x_src[5] and W[5,128]
__global__ void gat_edge_agg_fly_kernel(const int* __restrict__ src, const int* __restrict__ dst,
                                        const float* __restrict__ ew, const float* __restrict__ cep,
                                        const float* __restrict__ ssc, const float* __restrict__ dsc,
                                        const unsigned* __restrict__ menc, const float* __restrict__ den,
                                        const float* __restrict__ xsrc, const float* __restrict__ W,
                                        float* __restrict__ out, int E) {
  int gid = blockIdx.x * blockDim.x + threadIdx.x;
  int e = gid >> 5;
  if (e >= E) return;
  int lane = gid & 31;
  int s = src[e], d = dst[e];
  float alpha = edge_alpha(s, d, ew[e], cep[0], ssc, dsc, menc, den);
  const float* xp = xsrc + (size_t)s * 5;
  float x0 = xp[0], x1 = xp[1], x2 = xp[2], x3 = xp[3], x4 = xp[4];
  float* op = out + (size_t)d * CDIM;
  int c0 = lane * 4;
#pragma unroll
  for (int j = 0; j < 4; ++j) {
    int c = c0 + j;
    float h = x0 * W[c] + x1 * W[128 + c] + x2 * W[256 + c] + x3 * W[384 + c] + x4 * W[512 + c];
    atomicAdd(op + c, alpha * h);
  }
}

// ---------------- SAGE aggregation (sum + count) ----------------
__global__ void sage_agg_kernel(const int* __restrict__ src, const int* __restrict__ dst,
                                const float* __restrict__ x, int F,
                                float* __restrict__ agg, float* __restrict__ cnt, int E) {
  int gid = blockIdx.x * blockDim.x + threadIdx.x;
  int e = gid >> 5;
  if (e >= E) return;
  int lane = gid & 31;
  int s = src[e], d = dst[e];
  int per = F >> 5;  // 1 (F=32), 2 (F=64), 4 (F=128)
  const float* xp = x + (size_t)s * F + lane * per;
  float* ap = agg + (size_t)d * F + lane * per;
  if (per == 4) {
    float4 v = *(const float4*)xp;
    atomicAdd(ap + 0, v.x);
    atomicAdd(ap + 1, v.y);
    atomicAdd(ap + 2, v.z);
    atomicAdd(ap + 3, v.w);
  } else if (per == 2) {
    float2 v = *(const float2*)xp;
    atomicAdd(ap + 0, v.x);
    atomicAdd(ap + 1, v.y);
  } else {
    atomicAdd(ap, xp[0]);
  }
  if (lane == 0) atomicAdd(cnt + d, 1.f);
}

// ---------------- LayerNorm + ReLU, in place, wave per row ----------------
__global__ void ln_relu_kernel(float* __restrict__ xio, const float* __restrict__ g,
                               const float* __restrict__ b, int N) {
  int gid = blockIdx.x * blockDim.x + threadIdx.x;
  int row = gid >> 5;
  if (row >= N) return;
  int lane = gid & 31;
  float* rp = xio + (size_t)row * CDIM;
  float4 v = *(const float4*)(rp + lane * 4);
  float s = v.x + v.y + v.z + v.w;
#pragma unroll
  for (int o = 16; o > 0; o >>= 1) s += __shfl_xor(s, o, 32);
  float mu = s * 0.0078125f;
  float dx = v.x - mu, dy = v.y - mu, dz = v.z - mu, dw = v.w - mu;
  float q = dx * dx + dy * dy + dz * dz + dw * dw;
#pragma unroll
  for (int o = 16; o > 0; o >>= 1) q += __shfl_xor(q, o, 32);
  float rstd = rsqrtf(q * 0.0078125f + 1e-5f);
  int c = lane * 4;
  float4 r;
  r.x = fmaxf(dx * rstd * g[c + 0] + b[c + 0], 0.f);
  r.y = fmaxf(dy * rstd * g[c + 1] + b[c + 1], 0.f);
  r.z = fmaxf(dz * rstd * g[c + 2] + b[c + 2], 0.f);
  r.w = fmaxf(dw * rstd * g[c + 3] + b[c + 3], 0.f);
  *(float4*)(rp + c) = r;
}

// ---------------- fp32 WMMA GEMM combine ----------------
// One 8-wave block computes a 16-row x 128-col tile of D.
// D = sum_i rowscale_i * (A_i @ B_i)  + base + bias, optional ReLU.
// A: [M,K] row-major; B: [K,128] row-major; rowscale = 1/max(cnt,1) when cnt != null.
//
// Out-of-range A rows are CLAMPED to row M-1 (not predicated): each output row
// depends only on its own A row, and rows >= M are masked at the store, so the
// duplicated rows are harmless. This keeps EXEC full and the loads branch-free.
__device__ __forceinline__ v8f gemm_tile(const float* __restrict__ A, const float* __restrict__ B,
                                         int K, int M, int row0, int l16, int half, int n) {
  v8f t = {};
  int ar = row0 + l16;
  int arc = ar < M ? ar : (M - 1);
  const float* arow = A + (size_t)arc * K;
  const float* bcol = B + n;
#pragma unroll 8
  for (int k = 0; k < K; k += 4) {
    int ka = k + 2 * half;                  // even -> 8B aligned
    v2f a = *(const v2f*)(arow + ka);       // A frag: lane l -> row l%16, K = ka,ka+1
    v2f b;
    b.x = bcol[(size_t)ka * CDIM];          // B frag: lane l -> col n, K = ka,ka+1
    b.y = bcol[(size_t)(ka + 1) * CDIM];
    t = __builtin_amdgcn_wmma_f32_16x16x4_f32(false, a, false, b, (short)0, t, false, false);
  }
  return t;
}

__device__ __forceinline__ void add_term(v8f& acc, v8f t, const float* cnt, int row0, int half, int M) {
#pragma unroll
  for (int v = 0; v < 8; ++v) {
    float s = 1.f;
    if (cnt) {
      int r = row0 + v + 8 * half;
      s = (r < M) ? (1.f / fmaxf(cnt[r], 1.f)) : 0.f;
    }
    acc[v] += t[v] * s;
  }
}

__global__ __launch_bounds__(256) void wmma_combine_kernel(
    float* __restrict__ out, int M,
    const float* __restrict__ base, const float* __restrict__ bias,
    const float* A0, const float* B0, int K0, const float* cnt0,
    const float* A1, const float* B1, int K1, const float* cnt1,
    const float* A2, const float* B2, int K2, const float* cnt2,
    int do_relu) {
  int lane = threadIdx.x & 31;
  int wave = threadIdx.x >> 5;
  int half = lane >> 4;
  int l16 = lane & 15;
  int row0 = blockIdx.x * 16;
  int n = wave * 16 + l16;  // column 0..127
  v8f acc = {};
  if (A0) add_term(acc, gemm_tile(A0, B0, K0, M, row0, l16, half, n), cnt0, row0, half, M);
  if (A1) add_term(acc, gemm_tile(A1, B1, K1, M, row0, l16, half, n), cnt1, row0, half, M);
  if (A2) add_term(acc, gemm_tile(A2, B2, K2, M, row0, l16, half, n), cnt2, row0, half, M);
  float bv = bias ? bias[n] : 0.f;
#pragma unroll
  for (int v = 0; v < 8; ++v) {
    int r = row0 + v + 8 * half;  // C/D layout: VGPR v, halfwave -> row v + 8*half
    if (r < M) {
      float val = acc[v] + bv;
      if (base) val += base[(size_t)r * CDIM + n];
      if (do_relu) val = fmaxf(val, 0.f);
      out[(size_t)r * CDIM + n] = val;
    }
  }
}

// ---------------- host-side orchestration ----------------
enum {
  IN_XS = 0, IN_XC, IN_XK,
  IN_EW_ENR, IN_EW_ENG, IN_EW_RENR, IN_EW_RENG,
  IN_GE_WS, IN_GE_WD, IN_GE_WE, IN_GE_AS, IN_GE_AD, IN_GE_AE, IN_GE_B,          // gat_enr
  IN_GG_WS, IN_GG_WD, IN_GG_WE, IN_GG_AS, IN_GG_AD, IN_GG_AE, IN_GG_B,          // gat_eng
  IN_GR_WS, IN_GR_WD, IN_GR_WE, IN_GR_AS, IN_GR_AD, IN_GR_AE, IN_GR_B,          // gat_renr
  IN_GQ_WS, IN_GQ_WD, IN_GQ_WE, IN_GQ_AS, IN_GQ_AD, IN_GQ_AE, IN_GQ_B,          // gat_reng
  IN_SP_WL, IN_SP_BL, IN_SP_WR,                                                  // sage_prq
  IN_SC_WL, IN_SC_BL, IN_SC_WR,                                                  // sage_cov
  IN_SK_WL, IN_SK_BL, IN_SK_WR,                                                  // sage_cpr
  IN_S2R_WL, IN_S2R_BL, IN_S2R_WR,                                               // sage2_renr
  IN_S2Q_WL, IN_S2Q_BL, IN_S2Q_WR,                                               // sage2_reng
  IN_PROJ_W, IN_PROJ_B, IN_LN_G, IN_LN_B,
  IN_EI_ENR, IN_EI_ENG, IN_EI_PRQ, IN_EI_COV, IN_EI_CPR, IN_EI_RENR, IN_EI_RENG
};

static inline int cdiv(long a, long b) { return (int)((a + b - 1) / b); }

extern "C" void kernel_launch(void* const* d_in, const int* in_sizes, int n_in,
                              void* d_out, int out_size, void* d_ws, size_t ws_size,
                              hipStream_t stream) {
  (void)in_sizes; (void)n_in; (void)out_size; (void)ws_size;
  const float* F[64];
  for (int i = 0; i < 61; ++i) F[i] = (const float*)d_in[i];

  const float* x_student = F[IN_XS];
  const float* x_course  = F[IN_XC];
  const float* x_concept = F[IN_XK];

  const int* ei_enr  = (const int*)d_in[IN_EI_ENR];
  const int* ei_eng  = (const int*)d_in[IN_EI_ENG];
  const int* ei_prq  = (const int*)d_in[IN_EI_PRQ];
  const int* ei_cov  = (const int*)d_in[IN_EI_COV];
  const int* ei_cpr  = (const int*)d_in[IN_EI_CPR];
  const int* ei_renr = (const int*)d_in[IN_EI_RENR];
  const int* ei_reng = (const int*)d_in[IN_EI_RENG];

  // ---- workspace carve-up (floats) ----
  float* W = (float*)d_ws;
  size_t off = 0;
  auto A = [&](size_t nfl) { float* p = W + off; off += (nfl + 255) & ~(size_t)255; return p; };
  float* sp        = A(SP_TOTAL);
  float* wr_sumK   = A(32 * 128);    // Wr_cov + Wr_cpr
  float* wr_sum2   = A(128 * 128);   // Wr2_renr + Wr2_reng
  float* ssc_enr   = A(NS);  float* ssc_eng  = A(NS);
  float* dsc_renr  = A(NS);  float* dsc_reng = A(NS);
  float* ssc_renr  = A(NC);  float* dsc_enr  = A(NC);
  float* ssc_reng  = A(NK);  float* dsc_eng  = A(NK);
  unsigned* m_enr  = (unsigned*)A(NC);  float* den_enr  = A(NC);
  unsigned* m_eng  = (unsigned*)A(NK);  float* den_eng  = A(NK);
  unsigned* m_renr = (unsigned*)A(NS);  float* den_renr = A(NS);
  unsigned* m_reng = (unsigned*)A(NS);  float* den_reng = A(NS);
  float* hs_renr   = A((size_t)NC * CDIM);  // x_course  @ Ws_renr
  float* hs_reng   = A((size_t)NK * CDIM);  // x_concept @ Ws_reng
  float* c_fs      = A((size_t)NC * CDIM);
  float* k_fs      = A((size_t)NK * CDIM);
  float* s_acc     = A((size_t)NS * CDIM);  // becomes x_student_1 after LN (in place)
  float* agg_prq   = A((size_t)NC * 64);  float* cnt_prq  = A(NC);
  float* agg_cov   = A((size_t)NK * 64);  float* cnt_cov  = A(NK);
  float* agg_cpr   = A((size_t)NK * 32);  float* cnt_cpr  = A(NK);
  float* agg2_renr = A((size_t)NS * CDIM); float* cnt2_renr = A(NS);
  float* agg2_reng = A((size_t)NS * CDIM); float* cnt2_reng = A(NS);
  size_t used_bytes = off * sizeof(float);
  hipMemsetAsync(d_ws, 0, used_bytes, stream);

  float* out_student = (float*)d_out;
  float* out_course  = out_student + (size_t)NS * CDIM;
  float* out_concept = out_course + (size_t)NC * CDIM;

  // ---- tiny derived params ----
  PrepArgs pa;
  pa.Ws_enr = F[IN_GE_WS]; pa.as_enr = F[IN_GE_AS];
  pa.Ws_eng = F[IN_GG_WS]; pa.as_eng = F[IN_GG_AS];
  pa.Wd_renr = F[IN_GR_WD]; pa.ad_renr = F[IN_GR_AD];
  pa.Wd_reng = F[IN_GQ_WD]; pa.ad_reng = F[IN_GQ_AD];
  pa.Ws_renr = F[IN_GR_WS]; pa.as_renr = F[IN_GR_AS];
  pa.Wd_enr = F[IN_GE_WD]; pa.ad_enr = F[IN_GE_AD];
  pa.Ws_reng = F[IN_GQ_WS]; pa.as_reng = F[IN_GQ_AS];
  pa.Wd_eng = F[IN_GG_WD]; pa.ad_eng = F[IN_GG_AD];
  pa.We_enr = F[IN_GE_WE]; pa.ae_enr = F[IN_GE_AE];
  pa.We_eng = F[IN_GG_WE]; pa.ae_eng = F[IN_GG_AE];
  pa.We_renr = F[IN_GR_WE]; pa.ae_renr = F[IN_GR_AE];
  pa.We_reng = F[IN_GQ_WE]; pa.ae_reng = F[IN_GQ_AE];
  pa.b_enr = F[IN_GE_B]; pa.b_eng = F[IN_GG_B]; pa.b_renr = F[IN_GR_B]; pa.b_reng = F[IN_GQ_B];
  pa.bl_prq = F[IN_SP_BL]; pa.bl_cov = F[IN_SC_BL]; pa.bl_cpr = F[IN_SK_BL];
  pa.bl2_renr = F[IN_S2R_BL]; pa.bl2_reng = F[IN_S2Q_BL]; pa.proj_b = F[IN_PROJ_B];
  prep_vectors_kernel<<<1, 256, 0, stream>>>(pa, sp);
  add_arrays_kernel<<<cdiv(32 * 128, 256), 256, 0, stream>>>(F[IN_SC_WR], F[IN_SK_WR], wr_sumK, 32 * 128);
  add_arrays_kernel<<<cdiv(128 * 128, 256), 256, 0, stream>>>(F[IN_S2R_WR], F[IN_S2Q_WR], wr_sum2, 128 * 128);

  // ---- per-node attention scalars ----
  node_scores_kernel<<<cdiv(NS, 256), 256, 0, stream>>>(x_student, 5, sp + SP_WSAS_ENR, sp + SP_WSAS_ENG, ssc_enr, ssc_eng, NS);
  node_scores_kernel<<<cdiv(NS, 256), 256, 0, stream>>>(x_student, 5, sp + SP_WDAD_RENR, sp + SP_WDAD_RENG, dsc_renr, dsc_reng, NS);
  node_scores_kernel<<<cdiv(NC, 256), 256, 0, stream>>>(x_course, 64, sp + SP_WSAS_RENR, sp + SP_WDAD_ENR, ssc_renr, dsc_enr, NC);
  node_scores_kernel<<<cdiv(NK, 256), 256, 0, stream>>>(x_concept, 32, sp + SP_WSAS_RENG, sp + SP_WDAD_ENG, ssc_reng, dsc_eng, NK);

  // ---- hs tables for course/concept-sourced GATs (WMMA GEMMs) ----
  wmma_combine_kernel<<<cdiv(NC, 16), 256, 0, stream>>>(hs_renr, NC, nullptr, nullptr,
      x_course, F[IN_GR_WS], 64, nullptr, nullptr, nullptr, 0, nullptr, nullptr, nullptr, 0, nullptr, 0);
  wmma_combine_kernel<<<cdiv(NK, 16), 256, 0, stream>>>(hs_reng, NK, nullptr, nullptr,
      x_concept, F[IN_GQ_WS], 32, nullptr, nullptr, nullptr, 0, nullptr, nullptr, nullptr, 0, nullptr, 0);

  // ---- student skip init (must precede renr/reng scatters into s_acc) ----
  s_skip_kernel<<<cdiv((long)NS * CDIM, 256), 256, 0, stream>>>(x_student, F[IN_PROJ_W], sp + SP_SBIAS, s_acc, NS);

  // ---- GAT enr: student -> course (hs on the fly) ----
  {
    const int *s = ei_enr, *d = ei_enr + E_ENR; const float* ew = F[IN_EW_ENR]; const float* ce = sp + SP_CE + 0;
    gat_edge_max_kernel<<<cdiv(E_ENR, 256), 256, 0, stream>>>(s, d, ew, ce, ssc_enr, dsc_enr, m_enr, E_ENR);
    gat_edge_den_kernel<<<cdiv(E_ENR, 256), 256, 0, stream>>>(s, d, ew, ce, ssc_enr, dsc_enr, m_enr, den_enr, E_ENR);
    gat_edge_agg_fly_kernel<<<cdiv((long)E_ENR * 32, 256), 256, 0, stream>>>(s, d, ew, ce, ssc_enr, dsc_enr, m_enr, den_enr, x_student, F[IN_GE_WS], c_fs, E_ENR);
  }
  // ---- GAT eng: student -> concept ----
  {
    const int *s = ei_eng, *d = ei_eng + E_ENG; const float* ew = F[IN_EW_ENG]; const float* ce = sp + SP_CE + 1;
    gat_edge_max_kernel<<<cdiv(E_ENG, 256), 256, 0, stream>>>(s, d, ew, ce, ssc_eng, dsc_eng, m_eng, E_ENG);
    gat_edge_den_kernel<<<cdiv(E_ENG, 256), 256, 0, stream>>>(s, d, ew, ce, ssc_eng, dsc_eng, m_eng, den_eng, E_ENG);
    gat_edge_agg_fly_kernel<<<cdiv((long)E_ENG * 32, 256), 256, 0, stream>>>(s, d, ew, ce, ssc_eng, dsc_eng, m_eng, den_eng, x_student, F[IN_GG_WS], k_fs, E_ENG);
  }
  // ---- GAT renr: course -> student (materialized hs) ----
  {
    const int *s = ei_renr, *d = ei_renr + E_ENR; const float* ew = F[IN_EW_RENR]; const float* ce = sp + SP_CE + 2;
    gat_edge_max_kernel<<<cdiv(E_ENR, 256), 256, 0, stream>>>(s, d, ew, ce, ssc_renr, dsc_renr, m_renr, E_ENR);
    gat_edge_den_kernel<<<cdiv(E_ENR, 256), 256, 0, stream>>>(s, d, ew, ce, ssc_renr, dsc_renr, m_renr, den_renr, E_ENR);
    gat_edge_agg_mat_kernel<<<cdiv((long)E_ENR * 32, 256), 256, 0, stream>>>(s, d, ew, ce, ssc_renr, dsc_renr, m_renr, den_renr, hs_renr, s_acc, E_ENR);
  }
  // ---- GAT reng: concept -> student ----
  {
    const int *s = ei_reng, *d = ei_reng + E_ENG; const float* ew = F[IN_EW_RENG]; const float* ce = sp + SP_CE + 3;
    gat_edge_max_kernel<<<cdiv(E_ENG, 256), 256, 0, stream>>>(s, d, ew, ce, ssc_reng, dsc_reng, m_reng, E_ENG);
    gat_edge_den_kernel<<<cdiv(E_ENG, 256), 256, 0, stream>>>(s, d, ew, ce, ssc_reng, dsc_reng, m_reng, den_reng, E_ENG);
    gat_edge_agg_mat_kernel<<<cdiv((long)E_ENG * 32, 256), 256, 0, stream>>>(s, d, ew, ce, ssc_reng, dsc_reng, m_reng, den_reng, hs_reng, s_acc, E_ENG);
  }

  // ---- SAGE layer-1 aggregations ----
  sage_agg_kernel<<<cdiv((long)E_PRQ * 32, 256), 256, 0, stream>>>(ei_prq, ei_prq + E_PRQ, x_course, 64, agg_prq, cnt_prq, E_PRQ);
  sage_agg_kernel<<<cdiv((long)E_COV * 32, 256), 256, 0, stream>>>(ei_cov, ei_cov + E_COV, x_course, 64, agg_cov, cnt_cov, E_COV);
  sage_agg_kernel<<<cdiv((long)E_CPR * 32, 256), 256, 0, stream>>>(ei_cpr, ei_cpr + E_CPR, x_concept, 32, agg_cpr, cnt_cpr, E_CPR);

  // ---- x_course_1 = relu(c_fs + mean_prq@Wl + x_course@Wr + bias) ----
  wmma_combine_kernel<<<cdiv(NC, 16), 256, 0, stream>>>(out_course, NC, c_fs, sp + SP_BIAS_COURSE,
      agg_prq, F[IN_SP_WL], 64, cnt_prq,
      x_course, F[IN_SP_WR], 64, nullptr,
      nullptr, nullptr, 0, nullptr, 1);
  // ---- x_concept_1 = relu(k_fs + mean_cov@Wl_cov + mean_cpr@Wl_cpr + x_concept@(Wr_cov+Wr_cpr) + bias) ----
  wmma_combine_kernel<<<cdiv(NK, 16), 256, 0, stream>>>(out_concept, NK, k_fs, sp + SP_BIAS_CONCEPT,
      agg_cov, F[IN_SC_WL], 64, cnt_cov,
      agg_cpr, F[IN_SK_WL], 32, cnt_cpr,
      x_concept, wr_sumK, 32, nullptr, 1);

  // ---- x_student_1 = relu(LN(s_acc)), in place ----
  ln_relu_kernel<<<cdiv((long)NS * 32, 256), 256, 0, stream>>>(s_acc, F[IN_LN_G], F[IN_LN_B], NS);

  // ---- SAGE layer-2 aggregations (sources are layer-1 outputs) ----
  sage_agg_kernel<<<cdiv((long)E_ENR * 32, 256), 256, 0, stream>>>(ei_renr, ei_renr + E_ENR, out_course, 128, agg2_renr, cnt2_renr, E_ENR);
  sage_agg_kernel<<<cdiv((long)E_ENG * 32, 256), 256, 0, stream>>>(ei_reng, ei_reng + E_ENG, out_concept, 128, agg2_reng, cnt2_reng, E_ENG);

  // ---- x_student_out = relu(mean2_renr@Wl + mean2_reng@Wl + x_s1@(Wr+Wr) + bias) ----
  wmma_combine_kernel<<<cdiv(NS, 16), 256, 0, stream>>>(out_student, NS, nullptr, sp + SP_BIAS_SOUT,
      agg2_renr, F[IN_S2R_WL], 128, cnt2_renr,
      agg2_reng, F[IN_S2Q_WL], 128, cnt2_reng,
      s_acc, wr_sum2, 128, nullptr, 1);
}